// CapOnlyContrastiveLoss_56229711839260
// MI455X (gfx1250) — compile-verified
//
#include <hip/hip_runtime.h>
#include <math.h>

typedef __attribute__((ext_vector_type(2))) float v2f;
typedef __attribute__((ext_vector_type(8))) float v8f;
typedef __attribute__((ext_vector_type(4))) int   v4i;

#define NROWS 8192
#define DIMK  512
#define LMARGIN 0.2f

// GEMM tiling: block tile 128x128, K-chunk 32, wave tile 32x64 (2x4 wmma accs)
#define BM 128
#define BN 128
#define BK 32
#define LSTRIDE 36   // padded LDS row stride (floats); 144B -> 16B-aligned b128, 8B-aligned b64

#define AS1 __attribute__((address_space(1)))
#define AS3 __attribute__((address_space(3)))

#if defined(__has_builtin)
#if __has_builtin(__builtin_amdgcn_global_load_async_to_lds_b128) && \
    __has_builtin(__builtin_amdgcn_s_wait_asynccnt)
#define USE_ASYNC_LDS 1
#endif
#endif
#ifndef USE_ASYNC_LDS
#define USE_ASYNC_LDS 0
#endif

// ---------------------------------------------------------------------------
// Pass 1: per-row scalars.
//   c[i]    = MARGIN + ||im_i - s_i||
//   imsq[i] = ||im_i||^2 ; exsq[i] = ||ex_i||^2
// ---------------------------------------------------------------------------
__global__ __launch_bounds__(256) void prep_kernel(
    const float* __restrict__ im, const float* __restrict__ s,
    const float* __restrict__ ex,
    float* __restrict__ c_out, float* __restrict__ imsq, float* __restrict__ exsq)
{
    const int row  = blockIdx.x * 8 + (threadIdx.x >> 5);
    const int lane = threadIdx.x & 31;
    const float* imr = im + (size_t)row * DIMK;
    const float* sr  = s  + (size_t)row * DIMK;
    const float* exr = ex + (size_t)row * DIMK;

    float a2 = 0.f, df2 = 0.f, e2 = 0.f;
    for (int d = lane; d < DIMK; d += 32) {
        float a = imr[d], b = sr[d], e = exr[d];
        a2 += a * a;
        float df = a - b;
        df2 += df * df;
        e2 += e * e;
    }
    #pragma unroll
    for (int off = 16; off > 0; off >>= 1) {
        a2  += __shfl_xor(a2,  off, 32);
        df2 += __shfl_xor(df2, off, 32);
        e2  += __shfl_xor(e2,  off, 32);
    }
    if (lane == 0) {
        c_out[row] = LMARGIN + __builtin_sqrtf(df2);
        imsq[row]  = a2;
        exsq[row]  = e2;
    }
}

// ---------------------------------------------------------------------------
// Pass 2: fused GEMM (V_WMMA_F32_16X16X4_F32) + hinge epilogue + block partial.
// 256 threads = 8 waves; wave grid 4(M) x 2(N); wave tile 32x64.
// Double-buffered LDS; async global->LDS (ASYNCcnt) when the builtin exists.
// ---------------------------------------------------------------------------
__global__ __launch_bounds__(256) void gemm_loss_kernel(
    const float* __restrict__ im, const float* __restrict__ ex,
    const float* __restrict__ c_arr, const float* __restrict__ imsq,
    const float* __restrict__ exsq, float* __restrict__ partials)
{
    __shared__ __align__(16) float At[2][BM * LSTRIDE];
    __shared__ __align__(16) float Bt[2][BN * LSTRIDE];
    __shared__ float crow[BM];
    __shared__ float isq[BM];
    __shared__ float esq[BN];
    __shared__ float wsum[8];

    const int t      = threadIdx.x;
    const int m_base = blockIdx.y * BM;
    const int n_base = blockIdx.x * BN;

    if (t < BM) {
        crow[t] = c_arr[m_base + t];
        isq[t]  = imsq[m_base + t];
        esq[t]  = exsq[n_base + t];
    }

    const int wave   = t >> 5;
    const int lane   = t & 31;
    const int laneLo = lane & 15;
    const int laneHi = lane >> 4;
    const int wm = (wave & 3) * 32;    // wave M offset: 0,32,64,96
    const int wn = (wave >> 2) * 64;   // wave N offset: 0,64

    // Fragment row offsets in LDS (floats)
    const int aRow0 = (wm + laneLo)      * LSTRIDE;
    const int aRow1 = (wm + 16 + laneLo) * LSTRIDE;
    const int bRow0 = (wn + 0  + laneLo) * LSTRIDE;
    const int bRow1 = (wn + 16 + laneLo) * LSTRIDE;
    const int bRow2 = (wn + 32 + laneLo) * LSTRIDE;
    const int bRow3 = (wn + 48 + laneLo) * LSTRIDE;

    const v8f vzero = {0.f, 0.f, 0.f, 0.f, 0.f, 0.f, 0.f, 0.f};
    v8f acc[2][4];
    #pragma unroll
    for (int i = 0; i < 2; i++)
        #pragma unroll
        for (int j = 0; j < 4; j++) acc[i][j] = vzero;

    // Staging assignment: thread t covers rows r0, r0+32, r0+64, r0+96; 4 floats at c4.
    const int r0 = t >> 3;          // 0..31
    const int c4 = (t & 7) * 4;     // 0..28

    const int NCHUNK = DIMK / BK;   // 16

#if USE_ASYNC_LDS
    // ---- async global -> LDS pipeline (one barrier per chunk) ----
    #define ISSUE_CHUNK(KC, BUF)                                                        \
        do {                                                                            \
            _Pragma("unroll")                                                           \
            for (int rr = 0; rr < 4; rr++) {                                            \
                int row = r0 + rr * 32;                                                 \
                __builtin_amdgcn_global_load_async_to_lds_b128(                         \
                    (AS1 v4i*)(im + (size_t)(m_base + row) * DIMK + (KC) + c4),         \
                    (AS3 v4i*)&At[BUF][row * LSTRIDE + c4], 0, 0);                      \
                __builtin_amdgcn_global_load_async_to_lds_b128(                         \
                    (AS1 v4i*)(ex + (size_t)(n_base + row) * DIMK + (KC) + c4),         \
                    (AS3 v4i*)&Bt[BUF][row * LSTRIDE + c4], 0, 0);                      \
            }                                                                           \
        } while (0)

    ISSUE_CHUNK(0, 0);
    for (int kc = 0; kc < NCHUNK; ++kc) {
        const int cur = kc & 1;
        __builtin_amdgcn_s_wait_asynccnt(0);
        __syncthreads();
        if (kc + 1 < NCHUNK) ISSUE_CHUNK((kc + 1) * BK, cur ^ 1);

        const float* Ab = &At[cur][0];
        const float* Bb = &Bt[cur][0];
        #pragma unroll
        for (int k = 0; k < BK; k += 4) {
            const int kk = k + 2 * laneHi;
            v2f a0 = *(const v2f*)(Ab + aRow0 + kk);
            v2f a1 = *(const v2f*)(Ab + aRow1 + kk);
            v2f b0 = *(const v2f*)(Bb + bRow0 + kk);
            v2f b1 = *(const v2f*)(Bb + bRow1 + kk);
            v2f b2 = *(const v2f*)(Bb + bRow2 + kk);
            v2f b3 = *(const v2f*)(Bb + bRow3 + kk);
            acc[0][0] = __builtin_amdgcn_wmma_f32_16x16x4_f32(false, a0, false, b0, (short)0, acc[0][0], false, false);
            acc[0][1] = __builtin_amdgcn_wmma_f32_16x16x4_f32(false, a0, false, b1, (short)0, acc[0][1], false, false);
            acc[0][2] = __builtin_amdgcn_wmma_f32_16x16x4_f32(false, a0, false, b2, (short)0, acc[0][2], false, false);
            acc[0][3] = __builtin_amdgcn_wmma_f32_16x16x4_f32(false, a0, false, b3, (short)0, acc[0][3], false, false);
            acc[1][0] = __builtin_amdgcn_wmma_f32_16x16x4_f32(false, a1, false, b0, (short)0, acc[1][0], false, false);
            acc[1][1] = __builtin_amdgcn_wmma_f32_16x16x4_f32(false, a1, false, b1, (short)0, acc[1][1], false, false);
            acc[1][2] = __builtin_amdgcn_wmma_f32_16x16x4_f32(false, a1, false, b2, (short)0, acc[1][2], false, false);
            acc[1][3] = __builtin_amdgcn_wmma_f32_16x16x4_f32(false, a1, false, b3, (short)0, acc[1][3], false, false);
        }
    }
#else
    // ---- synchronous fallback: global->regs overlapped with compute ----
    float4 ra[4], rb[4];
    #pragma unroll
    for (int rr = 0; rr < 4; rr++) {
        int row = r0 + rr * 32;
        ra[rr] = *(const float4*)(im + (size_t)(m_base + row) * DIMK + c4);
        rb[rr] = *(const float4*)(ex + (size_t)(n_base + row) * DIMK + c4);
    }
    #pragma unroll
    for (int rr = 0; rr < 4; rr++) {
        int row = r0 + rr * 32;
        *(float4*)&At[0][row * LSTRIDE + c4] = ra[rr];
        *(float4*)&Bt[0][row * LSTRIDE + c4] = rb[rr];
    }
    for (int kc = 0; kc < NCHUNK; ++kc) {
        const int cur = kc & 1;
        __syncthreads();
        if (kc + 1 < NCHUNK) {
            const int k0 = (kc + 1) * BK;
            #pragma unroll
            for (int rr = 0; rr < 4; rr++) {
                int row = r0 + rr * 32;
                ra[rr] = *(const float4*)(im + (size_t)(m_base + row) * DIMK + k0 + c4);
                rb[rr] = *(const float4*)(ex + (size_t)(n_base + row) * DIMK + k0 + c4);
            }
        }
        const float* Ab = &At[cur][0];
        const float* Bb = &Bt[cur][0];
        #pragma unroll
        for (int k = 0; k < BK; k += 4) {
            const int kk = k + 2 * laneHi;
            v2f a0 = *(const v2f*)(Ab + aRow0 + kk);
            v2f a1 = *(const v2f*)(Ab + aRow1 + kk);
            v2f b0 = *(const v2f*)(Bb + bRow0 + kk);
            v2f b1 = *(const v2f*)(Bb + bRow1 + kk);
            v2f b2 = *(const v2f*)(Bb + bRow2 + kk);
            v2f b3 = *(const v2f*)(Bb + bRow3 + kk);
            acc[0][0] = __builtin_amdgcn_wmma_f32_16x16x4_f32(false, a0, false, b0, (short)0, acc[0][0], false, false);
            acc[0][1] = __builtin_amdgcn_wmma_f32_16x16x4_f32(false, a0, false, b1, (short)0, acc[0][1], false, false);
            acc[0][2] = __builtin_amdgcn_wmma_f32_16x16x4_f32(false, a0, false, b2, (short)0, acc[0][2], false, false);
            acc[0][3] = __builtin_amdgcn_wmma_f32_16x16x4_f32(false, a0, false, b3, (short)0, acc[0][3], false, false);
            acc[1][0] = __builtin_amdgcn_wmma_f32_16x16x4_f32(false, a1, false, b0, (short)0, acc[1][0], false, false);
            acc[1][1] = __builtin_amdgcn_wmma_f32_16x16x4_f32(false, a1, false, b1, (short)0, acc[1][1], false, false);
            acc[1][2] = __builtin_amdgcn_wmma_f32_16x16x4_f32(false, a1, false, b2, (short)0, acc[1][2], false, false);
            acc[1][3] = __builtin_amdgcn_wmma_f32_16x16x4_f32(false, a1, false, b3, (short)0, acc[1][3], false, false);
        }
        if (kc + 1 < NCHUNK) {
            const int nxt = cur ^ 1;
            #pragma unroll
            for (int rr = 0; rr < 4; rr++) {
                int row = r0 + rr * 32;
                *(float4*)&At[nxt][row * LSTRIDE + c4] = ra[rr];
                *(float4*)&Bt[nxt][row * LSTRIDE + c4] = rb[rr];
            }
        }
    }
#endif

    // Epilogue on accumulators.
    // C/D layout: VGPR v -> lanes 0-15: M=v, N=laneLo; lanes 16-31: M=v+8, N=laneLo.
    float sum = 0.f;
    #pragma unroll
    for (int mt = 0; mt < 2; mt++) {
        #pragma unroll
        for (int nt = 0; nt < 4; nt++) {
            const int jl = wn + nt * 16 + laneLo;
            const float ej = esq[jl];
            #pragma unroll
            for (int v = 0; v < 8; v++) {
                const int il = wm + mt * 16 + v + 8 * laneHi;
                float d2 = isq[il] + ej - 2.0f * acc[mt][nt][v];
                d2 = fmaxf(d2, 0.0f);
                float cost = crow[il] - __builtin_sqrtf(d2);
                sum += fmaxf(cost, 0.0f);
            }
        }
    }

    // Deterministic block reduction.
    #pragma unroll
    for (int off = 16; off > 0; off >>= 1)
        sum += __shfl_xor(sum, off, 32);
    if (lane == 0) wsum[wave] = sum;
    __syncthreads();
    if (t == 0) {
        float tot = 0.f;
        #pragma unroll
        for (int w = 0; w < 8; w++) tot += wsum[w];
        partials[blockIdx.y * gridDim.x + blockIdx.x] = tot;
    }
}

// ---------------------------------------------------------------------------
// Pass 3: sum partials -> mean.
// ---------------------------------------------------------------------------
__global__ __launch_bounds__(256) void final_reduce_kernel(
    const float* __restrict__ partials, int n, float* __restrict__ out, float scale)
{
    __shared__ float red[256];
    float s = 0.f;
    for (int i = threadIdx.x; i < n; i += 256) s += partials[i];
    red[threadIdx.x] = s;
    __syncthreads();
    #pragma unroll
    for (int off = 128; off > 0; off >>= 1) {
        if ((int)threadIdx.x < off) red[threadIdx.x] += red[threadIdx.x + off];
        __syncthreads();
    }
    if (threadIdx.x == 0) out[0] = red[0] * scale;
}

extern "C" void kernel_launch(void* const* d_in, const int* in_sizes, int n_in,
                              void* d_out, int out_size, void* d_ws, size_t ws_size,
                              hipStream_t stream)
{
    (void)in_sizes; (void)n_in; (void)out_size; (void)ws_size;
    const float* im = (const float*)d_in[0];
    const float* s  = (const float*)d_in[1];
    const float* ex = (const float*)d_in[2];

    float* ws       = (float*)d_ws;
    float* c_arr    = ws;                 // [NROWS]
    float* imsq     = ws + NROWS;         // [NROWS]
    float* exsq     = ws + 2 * NROWS;     // [NROWS]
    float* partials = ws + 3 * NROWS;     // [mtiles*ntiles]

    const int mtiles = NROWS / BM;   // 64
    const int ntiles = NROWS / BN;   // 64
    const int nPart  = mtiles * ntiles;

    prep_kernel<<<NROWS / 8, 256, 0, stream>>>(im, s, ex, c_arr, imsq, exsq);
    gemm_loss_kernel<<<dim3(ntiles, mtiles), 256, 0, stream>>>(im, ex, c_arr, imsq, exsq, partials);
    final_reduce_kernel<<<1, 256, 0, stream>>>(partials, nPart, (float*)d_out,
                                               1.0f / ((float)NROWS * (float)NROWS));
}